// ReactomeGNN_30485677867013
// MI455X (gfx1250) — compile-verified
//
#include <hip/hip_runtime.h>
#include <hip/hip_bf16.h>

#define N_GENES 9229
#define N_MOD   3
#define PROJ    32
#define HID     64
#define BATCH   64
#define MPAD    9232          // 577 * 16, padded row count for WMMA tiles
#define LDSP    68            // padded LDS row stride (floats) -> conflict-free transpose

typedef float v2f __attribute__((ext_vector_type(2)));
typedef float v8f __attribute__((ext_vector_type(8)));

// ---- workspace layout (float offsets) ----
#define OFF_DEG   0
#define OFF_NODES (OFF_DEG   + MPAD)            // nodes0: MPAD x 32
#define OFF_HZ    (OFF_NODES + MPAD * PROJ)     // hz:     MPAD x 64 (reused layer1+2)
#define OFF_AGG   (OFF_HZ    + MPAD * HID)      // agg:    MPAD x 64 (reused layer1+2)
#define OFF_Z1    (OFF_AGG   + MPAD * HID)      // z1:     MPAD x 64
#define OFF_POOL  (OFF_Z1    + MPAD * HID)      // pooled: 64 x 64
#define WS_FLOATS (OFF_POOL  + BATCH * HID)     // ~2.08M floats ~= 8.3 MB

// -------- init: deg=1 (self loop), zero nodes0 / agg / z1 / pooled --------
__global__ void k_init(float* __restrict__ ws) {
  long long stride = (long long)gridDim.x * blockDim.x;
  long long t0 = (long long)blockIdx.x * blockDim.x + threadIdx.x;
  for (long long i = t0; i < MPAD; i += stride) ws[OFF_DEG + i] = 1.0f;
  for (long long i = t0; i < (long long)MPAD * PROJ; i += stride) ws[OFF_NODES + i] = 0.0f;
  long long zn = (long long)(WS_FLOATS - OFF_AGG);
  for (long long i = t0; i < zn; i += stride) ws[OFF_AGG + i] = 0.0f;
}

// -------- degree: deg[dst] += 1 per edge (duplicates accumulate) --------
__global__ void k_degree(const long long* __restrict__ dst, float* __restrict__ deg, int E) {
  int e = blockIdx.x * blockDim.x + threadIdx.x;
  if (e < E) atomicAdd(&deg[(int)dst[e]], 1.0f);
}

// -------- batch-0 modality fusion: nodes0 = relu(x0 @ Wf + bf) --------
__global__ void k_fuse0(const float* __restrict__ x, const float* __restrict__ Wf,
                        const float* __restrict__ bf, float* __restrict__ nodes0) {
  int t = blockIdx.x * blockDim.x + threadIdx.x;   // t = g*32 + p
  int g = t >> 5, p = t & 31;
  if (g >= N_GENES) return;
  const float* xg = x + g * 3;                     // batch 0
  float v = bf[p] + xg[0] * Wf[p] + xg[1] * Wf[PROJ + p] + xg[2] * Wf[2 * PROJ + p];
  nodes0[g * PROJ + p] = fmaxf(v, 0.0f);
}

// -------- WMMA fp32 GEMM: C[MPAD x 64] = A[MPAD x K] @ Bw[K x 64] --------
// One wave per 16x16 output tile; V_WMMA_F32_16X16X4_F32, K stepped by 4.
__global__ void k_gemm_wmma(const float* __restrict__ A, const float* __restrict__ Bw,
                            float* __restrict__ C, int K) {
  int wave = blockIdx.x * (blockDim.x >> 5) + (threadIdx.x >> 5);
  int lane = threadIdx.x & 31;
  const int tilesN = HID / 16;                 // 4
  const int tilesM = MPAD / 16;                // 577
  if (wave >= tilesM * tilesN) return;         // wave-uniform: EXEC stays all-1s
  int tm = wave / tilesN, tn = wave % tilesN;
  int ml    = lane & 15;                       // A row within tile / C column
  int khalf = lane >> 4;                       // 0: K=k,k+1  1: K=k+2,k+3
  const float* Arow = A + (long long)(tm * 16 + ml) * K;
  int n = tn * 16 + ml;                        // C/D column = lane&15
  v8f acc = {};
  for (int kk = 0; kk < K; kk += 4) {
    int kb = kk + khalf * 2;
    v2f a, b;
    a.x = Arow[kb + 0];                        // A 16x4 layout: VGPR0=K{0,2}, VGPR1=K{1,3}
    a.y = Arow[kb + 1];
    b.x = Bw[(kb + 0) * HID + n];              // B 4x16: rows striped across lane halves
    b.y = Bw[(kb + 1) * HID + n];
    acc = __builtin_amdgcn_wmma_f32_16x16x4_f32(
        /*neg_a=*/false, a, /*neg_b=*/false, b,
        /*c_mod=*/(short)0, acc, /*reuse_a=*/false, /*reuse_b=*/false);
  }
  // C/D layout: VGPR r -> row r (lanes 0-15) / row 8+r (lanes 16-31)
  float* Cp = C + (long long)(tm * 16 + khalf * 8) * HID + n;
  #pragma unroll
  for (int r = 0; r < 8; ++r) Cp[r * HID] = acc[r];
}

// -------- edge scatter: agg[dst] += hz[src] * rsqrt(deg[src])*rsqrt(deg[dst]) --------
__global__ void k_scatter(const long long* __restrict__ src, const long long* __restrict__ dst,
                          const float* __restrict__ deg, const float* __restrict__ hz,
                          float* __restrict__ agg, int E) {
  int t = blockIdx.x * blockDim.x + threadIdx.x;   // 4 threads/edge, 16 feats each
  int e = t >> 2;
  if (e >= E) return;
  int fo = (t & 3) * 16;
  int s = (int)src[e], d = (int)dst[e];
  float norm = rsqrtf(deg[s]) * rsqrtf(deg[d]);
  const float* hs = hz + (long long)s * HID + fo;
  float* ad = agg + (long long)d * HID + fo;
  #pragma unroll
  for (int f = 0; f < 16; ++f) atomicAdd(&ad[f], hs[f] * norm);
}

// -------- layer-1 epilogue: z1 = relu(agg + hz/deg + b1); reset agg for layer 2 --------
__global__ void k_post1(const float* __restrict__ hz, float* __restrict__ agg,
                        const float* __restrict__ deg, const float* __restrict__ b1,
                        float* __restrict__ z1) {
  int t = blockIdx.x * blockDim.x + threadIdx.x;   // t = g*64 + f
  int g = t >> 6, f = t & 63;
  if (g >= N_GENES) return;
  long long i = (long long)g * HID + f;
  float v = agg[i] + hz[i] / deg[g] + b1[f];
  z1[i] = fmaxf(v, 0.0f);
  agg[i] = 0.0f;                                   // re-zero for second scatter
}

// -------- layer-2 epilogue + mean-pool contribution for batch 0 --------
__global__ void k_post2_pool(const float* __restrict__ hz, const float* __restrict__ agg,
                             const float* __restrict__ deg, const float* __restrict__ b2,
                             float* __restrict__ pooled) {
  __shared__ float red[256];
  int f  = threadIdx.x & 63;
  int rs = threadIdx.x >> 6;                       // 0..3
  int base = blockIdx.x * 32;                      // 32 rows per block
  float s = 0.0f;
  #pragma unroll
  for (int i = 0; i < 8; ++i) {
    int g = base + rs + i * 4;
    if (g < N_GENES) {
      long long idx = (long long)g * HID + f;
      float v = agg[idx] + hz[idx] / deg[g] + b2[f];
      s += fmaxf(v, 0.0f);
    }
  }
  red[threadIdx.x] = s;
  __syncthreads();
  if (rs == 0) {
    float tot = red[f] + red[64 + f] + red[128 + f] + red[192 + f];
    atomicAdd(&pooled[f], tot);                    // batch 0 slot
  }
}

// -------- batches 1..63: edges never touch these nodes --------
// Fully fused WMMA MLP: one wave = 16 nodes. Layer-1 A fragments are produced
// directly in WMMA A-layout from the modality fusion; the D->A transpose between
// layers goes through padded LDS (stride 68 -> conflict-free). No thread exits
// before the last WMMA (EXEC all-1s requirement); invalid nodes are clamped on
// load and masked at pooling.
__global__ void __launch_bounds__(256)
k_rest_wmma(const float* __restrict__ x, const float* __restrict__ Wf,
            const float* __restrict__ bf, const float* __restrict__ W1,
            const float* __restrict__ b1, const float* __restrict__ W2,
            const float* __restrict__ b2, float* __restrict__ pooled) {
  __shared__ float pool[HID];
  __shared__ float h1s[8][16 * LDSP];              // 8 waves x 4.25KB = 34KB
  int tid = threadIdx.x;
  if (tid < HID) pool[tid] = 0.0f;
  __syncthreads();

  int wv    = tid >> 5;
  int lane  = tid & 31;
  int ml    = lane & 15;                           // node-in-tile (A) / column (B,C,D)
  int khalf = lane >> 4;                           // K-pair selector

  int b = blockIdx.y + 1;
  int nodeBase = blockIdx.x * 128 + wv * 16;       // 16 nodes per wave
  int g  = nodeBase + ml;
  int gg = g < N_GENES ? g : (N_GENES - 1);        // clamp; masked at pooling
  const float* xg = x + (long long)b * (N_GENES * N_MOD) + (long long)gg * 3;
  float x0 = xg[0], x1 = xg[1], x2 = xg[2];

  // --- modality fusion straight into layer-1 A fragments (16 x 32, fp32) ---
  v2f a1[8];
  #pragma unroll
  for (int i = 0; i < 8; ++i) {
    int j0 = 4 * i + 2 * khalf;
    a1[i].x = fmaxf(bf[j0]     + x0 * Wf[j0]     + x1 * Wf[PROJ + j0]     + x2 * Wf[2 * PROJ + j0],     0.0f);
    a1[i].y = fmaxf(bf[j0 + 1] + x0 * Wf[j0 + 1] + x1 * Wf[PROJ + j0 + 1] + x2 * Wf[2 * PROJ + j0 + 1], 0.0f);
  }

  // --- h1 = relu(A @ W1 + b1): 4 N-tiles x 8 K-steps; epilogue in D-layout ---
  float* hrow = &h1s[wv][0];
  #pragma unroll
  for (int t = 0; t < 4; ++t) {
    int n = t * 16 + ml;
    v8f acc = {};
    #pragma unroll
    for (int i = 0; i < 8; ++i) {
      int kb = 4 * i + 2 * khalf;
      v2f bv;
      bv.x = W1[kb * HID + n];
      bv.y = W1[(kb + 1) * HID + n];
      acc = __builtin_amdgcn_wmma_f32_16x16x4_f32(false, a1[i], false, bv,
                                                  (short)0, acc, false, false);
    }
    float bias = b1[n];
    #pragma unroll
    for (int r = 0; r < 8; ++r)                    // D row = khalf*8 + r, col = n
      hrow[(khalf * 8 + r) * LDSP + n] = fmaxf(acc[r] + bias, 0.0f);
  }

  // --- D-layout -> A-layout through LDS (same-wave RAW; dscnt handled) ---
  v2f a2[16];
  #pragma unroll
  for (int i = 0; i < 16; ++i) {
    int kb = 4 * i + 2 * khalf;
    a2[i].x = hrow[ml * LDSP + kb];
    a2[i].y = hrow[ml * LDSP + kb + 1];
  }

  // --- h2 = relu(h1 @ W2 + b2): 4 N-tiles x 16 K-steps; pool in D-layout ---
  #pragma unroll
  for (int t = 0; t < 4; ++t) {
    int n = t * 16 + ml;
    v8f acc = {};
    #pragma unroll
    for (int i = 0; i < 16; ++i) {
      int kb = 4 * i + 2 * khalf;
      v2f bv;
      bv.x = W2[kb * HID + n];
      bv.y = W2[(kb + 1) * HID + n];
      acc = __builtin_amdgcn_wmma_f32_16x16x4_f32(false, a2[i], false, bv,
                                                  (short)0, acc, false, false);
    }
    float bias = b2[n];
    float s = 0.0f;
    #pragma unroll
    for (int r = 0; r < 8; ++r) {                  // node = nodeBase + khalf*8 + r
      float v = fmaxf(acc[r] + bias, 0.0f);
      if (nodeBase + khalf * 8 + r < N_GENES) s += v;
    }
    s += __shfl_xor(s, 16, 32);                    // merge lane halves
    if (lane < 16) atomicAdd(&pool[t * 16 + lane], s);
  }
  __syncthreads();
  if (tid < HID) atomicAdd(&pooled[b * HID + tid], pool[tid]);
}

// -------- classifier: out[b] = (pooled[b]/G) . Wc + bc --------
__global__ void k_final(const float* __restrict__ pooled, const float* __restrict__ Wc,
                        const float* __restrict__ bc, float* __restrict__ out) {
  int b = threadIdx.x;
  if (b >= BATCH) return;
  float a = bc[0];
  const float inv = 1.0f / (float)N_GENES;
  #pragma unroll
  for (int k = 0; k < HID; ++k) a = fmaf(pooled[b * HID + k] * inv, Wc[k], a);
  out[b] = a;
}

extern "C" void kernel_launch(void* const* d_in, const int* in_sizes, int n_in,
                              void* d_out, int out_size, void* d_ws, size_t ws_size,
                              hipStream_t stream) {
  const float* x  = (const float*)d_in[0];
  const float* Wf = (const float*)d_in[1];
  const float* bf = (const float*)d_in[2];
  const float* W1 = (const float*)d_in[3];
  const float* b1 = (const float*)d_in[4];
  const float* W2 = (const float*)d_in[5];
  const float* b2 = (const float*)d_in[6];
  const float* Wc = (const float*)d_in[7];
  const float* bc = (const float*)d_in[8];
  const long long* ei = (const long long*)d_in[9];
  const int E = in_sizes[9] / 2;
  const long long* srcI = ei;
  const long long* dstI = ei + E;

  float* ws     = (float*)d_ws;
  float* deg    = ws + OFF_DEG;
  float* nodes0 = ws + OFF_NODES;
  float* hz     = ws + OFF_HZ;
  float* agg    = ws + OFF_AGG;
  float* z1     = ws + OFF_Z1;
  float* pooled = ws + OFF_POOL;
  float* out    = (float*)d_out;

  k_init<<<512, 256, 0, stream>>>(ws);
  k_degree<<<(E + 255) / 256, 256, 0, stream>>>(dstI, deg, E);
  k_fuse0<<<(N_GENES * PROJ + 255) / 256, 256, 0, stream>>>(x, Wf, bf, nodes0);
  // layer 1: hz = nodes0 @ W1  (K=32), scatter, epilogue
  k_gemm_wmma<<<MPAD / 16, 128, 0, stream>>>(nodes0, W1, hz, PROJ);
  k_scatter<<<(E * 4 + 255) / 256, 256, 0, stream>>>(srcI, dstI, deg, hz, agg, E);
  k_post1<<<(N_GENES * HID + 255) / 256, 256, 0, stream>>>(hz, agg, deg, b1, z1);
  // layer 2: hz = z1 @ W2  (K=64), scatter, epilogue + batch-0 pooling
  k_gemm_wmma<<<MPAD / 16, 128, 0, stream>>>(z1, W2, hz, HID);
  k_scatter<<<(E * 4 + 255) / 256, 256, 0, stream>>>(srcI, dstI, deg, hz, agg, E);
  k_post2_pool<<<(N_GENES + 31) / 32, 256, 0, stream>>>(hz, agg, deg, b2, pooled);
  // batches 1..63: fused WMMA dense path (graph never touches these nodes)
  dim3 gr((N_GENES + 127) / 128, BATCH - 1);
  k_rest_wmma<<<gr, 256, 0, stream>>>(x, Wf, bf, W1, b1, W2, b2, pooled);
  // classifier
  k_final<<<1, 64, 0, stream>>>(pooled, Wc, bc, out);
}